// LinEq2v2_14998025798530
// MI455X (gfx1250) — compile-verified
//
#include <hip/hip_runtime.h>

// Problem constants (reference: B=8, N=256, L=48, F=96)
#define BB 8
#define NN 256
#define LL 48
#define FF 96

typedef float v2f __attribute__((ext_vector_type(2)));
typedef float v8f __attribute__((ext_vector_type(8)));

// ---------------------------------------------------------------------------
// Kernel 1: rowsum[b,i,l] = sum_j x[b,i,j,l];  diag[b,i,l] = x[b,i,i,l]
// One block per (b,i). Row of x = 256*48 floats = 3072 float4, contiguous.
// Thread t: l4 = t%12 invariant (384 % 12 == 0); 8 float4 loads each.
// ---------------------------------------------------------------------------
__global__ __launch_bounds__(384) void reduce_row_kernel(
    const float* __restrict__ x, float* __restrict__ rowsum,
    float* __restrict__ diag) {
  const int blk = blockIdx.x;
  const int b = blk >> 8;
  const int i = blk & 255;
  const float* xrow = x + (size_t)(b * NN + i) * (NN * LL);
  const float4* xrow4 = (const float4*)xrow;
  const int t = threadIdx.x;  // 0..383
  float4 acc = make_float4(0.f, 0.f, 0.f, 0.f);
#pragma unroll
  for (int s = 0; s < 8; ++s) {
    float4 v = xrow4[t + 384 * s];
    acc.x += v.x; acc.y += v.y; acc.z += v.z; acc.w += v.w;
  }
  __shared__ float4 red[384];
  red[t] = acc;
  __syncthreads();
  const int o = (b * NN + i) * LL;
  if (t < 12) {
    float4 r = make_float4(0.f, 0.f, 0.f, 0.f);
#pragma unroll
    for (int g = 0; g < 32; ++g) {
      float4 v = red[t + 12 * g];
      r.x += v.x; r.y += v.y; r.z += v.z; r.w += v.w;
    }
    rowsum[o + 4 * t + 0] = r.x;
    rowsum[o + 4 * t + 1] = r.y;
    rowsum[o + 4 * t + 2] = r.z;
    rowsum[o + 4 * t + 3] = r.w;
  }
  if (t < LL) diag[o + t] = xrow[i * LL + t];
}

// ---------------------------------------------------------------------------
// Kernel 2: colsum[b,j,l] = sum_i x[b,i,j,l].  One block per (b,j).
// (x fits in 192MB L2, so this second pass over x is mostly L2 hits.)
// Thread t: l4 = t%12, ig = t/12 (0..31); i = ig + 32*s, s=0..7.
// ---------------------------------------------------------------------------
__global__ __launch_bounds__(384) void reduce_col_kernel(
    const float* __restrict__ x, float* __restrict__ colsum) {
  const int blk = blockIdx.x;
  const int b = blk >> 8;
  const int j = blk & 255;
  // float4 base of x[b, 0, j, 0]; row stride = 3072 float4
  const float4* base4 =
      (const float4*)(x + (size_t)b * NN * NN * LL + (size_t)j * LL);
  const int t = threadIdx.x;
  const int l4 = t % 12;
  const int ig = t / 12;  // 0..31
  float4 acc = make_float4(0.f, 0.f, 0.f, 0.f);
#pragma unroll
  for (int s = 0; s < 8; ++s) {
    const int i = ig + 32 * s;
    float4 v = base4[(size_t)i * (NN * LL / 4) + l4];
    acc.x += v.x; acc.y += v.y; acc.z += v.z; acc.w += v.w;
  }
  __shared__ float4 red[384];
  red[t] = acc;
  __syncthreads();
  if (t < 12) {
    float4 r = make_float4(0.f, 0.f, 0.f, 0.f);
#pragma unroll
    for (int g = 0; g < 32; ++g) {
      float4 v = red[t + 12 * g];
      r.x += v.x; r.y += v.y; r.z += v.z; r.w += v.w;
    }
    const int o = (b * NN + j) * LL;
    colsum[o + 4 * t + 0] = r.x;
    colsum[o + 4 * t + 1] = r.y;
    colsum[o + 4 * t + 2] = r.z;
    colsum[o + 4 * t + 3] = r.w;
  }
}

// ---------------------------------------------------------------------------
// Kernel 3: trace[b,l] = sum_i diag[b,i,l]; totsum[b,l] = sum_i rowsum[b,i,l]
// ---------------------------------------------------------------------------
__global__ __launch_bounds__(64) void trace_tot_kernel(
    const float* __restrict__ diag, const float* __restrict__ rowsum,
    float* __restrict__ trace, float* __restrict__ totsum) {
  const int b = blockIdx.x;
  const int t = threadIdx.x;
  if (t >= LL) return;
  float tr = 0.0f, ts = 0.0f;
  for (int i = 0; i < NN; ++i) {
    const int o = (b * NN + i) * LL + t;
    tr += diag[o];
    ts += rowsum[o];
  }
  trace[b * LL + t] = tr;
  totsum[b * LL + t] = ts;
}

// ---------------------------------------------------------------------------
// Kernel 4: small GEMMs -> bias tables.
//   bj[b,n,f]       = diag@w5 + rowsum@w6 + colsum@w7
//   bi2[b,n,f]      = diag@w8 + rowsum@w9 + colsum@w10 + trace@w13 + totsum@w14
//   diagterm[b,n,f] = diag@w0 + rowsum@w1 + colsum@w2 + trace@w3 + totsum@w4
// ---------------------------------------------------------------------------
__global__ __launch_bounds__(96) void combine_kernel(
    const float* __restrict__ rowsum, const float* __restrict__ colsum,
    const float* __restrict__ diag, const float* __restrict__ trace,
    const float* __restrict__ totsum, const float* __restrict__ w,
    float* __restrict__ bi2, float* __restrict__ bjv,
    float* __restrict__ dterm) {
  const int blk = blockIdx.x;
  const int b = blk >> 8;
  const int n = blk & 255;
  __shared__ float d[LL], rs[LL], cs[LL], tr[LL], ts[LL];
  const int t = threadIdx.x;
  if (t < LL) {
    const int base = (b * NN + n) * LL + t;
    d[t] = diag[base];
    rs[t] = rowsum[base];
    cs[t] = colsum[base];
    tr[t] = trace[b * LL + t];
    ts[t] = totsum[b * LL + t];
  }
  __syncthreads();
  const int f = t;  // 0..95
  float s_dn = 0.f, s_bj = 0.f, s_bi = 0.f;
  for (int l = 0; l < LL; ++l) {
    const float dv = d[l], rv = rs[l], cv = cs[l], tv = tr[l], sv = ts[l];
    const float* W = w + l * FF + f;
#define WK(k) W[(k) * (LL * FF)]
    s_dn += dv * WK(0) + rv * WK(1) + cv * WK(2) + tv * WK(3) + sv * WK(4);
    s_bj += dv * WK(5) + rv * WK(6) + cv * WK(7);
    s_bi += dv * WK(8) + rv * WK(9) + cv * WK(10) + tv * WK(13) + sv * WK(14);
#undef WK
  }
  const int ob = (b * NN + n) * FF + f;
  bjv[ob] = s_bj;
  bi2[ob] = s_bi;
  dterm[ob] = s_dn;
}

// ---------------------------------------------------------------------------
// Kernel 5 (main, WMMA): one 512-thread WG per (b, 16x16 (i,j) tile).
//   out[b,i,j,f] = x[b,i,j,:]@w12 + x[b,j,i,:]@w11
//                + bi2[b,i,f] + bj[b,j,f] + (i==j)*dterm[b,i,f]
// LDS pitch 52 dwords: 16B-aligned float4 staging (208 % 16 == 0) AND
// conflict-free ds_load_b64 fragments (52*m mod 64 distinct for m=0..15,
// and the two K-halves land on banks ==0/==2 mod 4 -> all 64 banks unique).
// 16 waves, one i'-strip each; 6 f-tiles; 12 k-steps x 2 terms of
// v_wmma_f32_16x16x4_f32. k-loop unrolled by 2 only, so B fragments stay
// transient (low VGPR count -> 4 waves/SIMD instead of 2).
// ---------------------------------------------------------------------------
#define XPITCH 52
#define XS_ELEMS (256 * XPITCH)  // 13312
#define WS_ELEMS (96 * XPITCH)   // 4992

__global__ __launch_bounds__(512) void main_wmma_kernel(
    const float* __restrict__ x, const float* __restrict__ w,
    const float* __restrict__ bi2, const float* __restrict__ bjv,
    const float* __restrict__ dterm, float* __restrict__ out) {
  extern __shared__ float smem[];
  float* Xs1 = smem;                             // 13312
  float* Xs2 = smem + XS_ELEMS;                  // 13312
  float* W12 = smem + 2 * XS_ELEMS;              // 4992
  float* W11 = smem + 2 * XS_ELEMS + WS_ELEMS;   // 4992

  const int b = blockIdx.z;
  const int i0 = blockIdx.y * 16;
  const int j0 = blockIdx.x * 16;
  const int tid = threadIdx.x;

  // Stage w11^T, w12^T: wT[f][l], pitch 52, so B fragments are float2.
  const float* w11g = w + 11 * (LL * FF);
  const float* w12g = w + 12 * (LL * FF);
  for (int e = tid; e < LL * FF; e += 512) {
    const int l = e / FF, f = e - l * FF;
    W12[f * XPITCH + l] = w12g[e];
    W11[f * XPITCH + l] = w11g[e];
  }

  const float4* x4 = (const float4*)x;
  const size_t xbase4 = (size_t)b * NN * NN * (LL / 4);
  // X1[(i'*16+j')*52 + 4*l4 ..] = x[b, i0+i', j0+j', 4*l4..] (b128 both sides)
#pragma unroll
  for (int s = 0; s < 6; ++s) {
    const int e4 = tid + 512 * s;  // 0..3071
    const int ip = e4 / 192;
    const int rem = e4 - ip * 192;
    const int jp = rem / 12;
    const int l4 = rem - jp * 12;
    float4 v = x4[xbase4 + ((size_t)(i0 + ip) * NN + (j0 + jp)) * 12 + l4];
    *(float4*)&Xs1[(ip * 16 + jp) * XPITCH + 4 * l4] = v;
  }
  // X2t[(i'*16+j')*52 + ..] = x[b, j0+j', i0+i', ..] (global reads contiguous)
#pragma unroll
  for (int s = 0; s < 6; ++s) {
    const int e4 = tid + 512 * s;
    const int jp = e4 / 192;
    const int rem = e4 - jp * 192;
    const int ip = rem / 12;
    const int l4 = rem - ip * 12;
    float4 v = x4[xbase4 + ((size_t)(j0 + jp) * NN + (i0 + ip)) * 12 + l4];
    *(float4*)&Xs2[(ip * 16 + jp) * XPITCH + 4 * l4] = v;
  }
  __syncthreads();

  const int wave = tid >> 5;  // 0..15 == i'-strip (mtile)
  const int lane = tid & 31;
  const int half = lane >> 4;  // 0: K 0..1 / rows 0..7 ; 1: K 2..3 / rows 8..15
  const int n16 = lane & 15;   // A row m / B,C column n

  const int mtile = wave;
  v8f C[6];
#pragma unroll
  for (int ft = 0; ft < 6; ++ft)
    C[ft] = (v8f){0.f, 0.f, 0.f, 0.f, 0.f, 0.f, 0.f, 0.f};

  const int arow = (mtile * 16 + n16) * XPITCH;
#pragma unroll 2
  for (int ks = 0; ks < 12; ++ks) {
    const int kk = ks * 4 + 2 * half;
    const v2f a1 = *(const v2f*)(Xs1 + arow + kk);  // identity term rows
    const v2f a2 = *(const v2f*)(Xs2 + arow + kk);  // transpose term rows
#pragma unroll
    for (int ft = 0; ft < 6; ++ft) {
      const v2f b12 = *(const v2f*)(W12 + (ft * 16 + n16) * XPITCH + kk);
      const v2f b11 = *(const v2f*)(W11 + (ft * 16 + n16) * XPITCH + kk);
      C[ft] = __builtin_amdgcn_wmma_f32_16x16x4_f32(
          false, a1, false, b12, (short)0, C[ft], false, false);
      C[ft] = __builtin_amdgcn_wmma_f32_16x16x4_f32(
          false, a2, false, b11, (short)0, C[ft], false, false);
    }
  }

  // Epilogue: C[ft] VGPR v holds out[b, i0+mtile, j0+v+8*half, ft*16+n16]
  const int i = i0 + mtile;
#pragma unroll
  for (int ft = 0; ft < 6; ++ft) {
    const int f = ft * 16 + n16;
    const float biv = bi2[(b * NN + i) * FF + f];
    const float dtv = dterm[(b * NN + i) * FF + f];
#pragma unroll
    for (int v = 0; v < 8; ++v) {
      const int j = j0 + v + 8 * half;
      float val = C[ft][v] + biv + bjv[(b * NN + j) * FF + f];
      if (i == j) val += dtv;
      out[((size_t)(b * NN + i) * NN + j) * FF + f] = val;
    }
  }
}

// ---------------------------------------------------------------------------
// Launch
// ---------------------------------------------------------------------------
extern "C" void kernel_launch(void* const* d_in, const int* in_sizes, int n_in,
                              void* d_out, int out_size, void* d_ws,
                              size_t ws_size, hipStream_t stream) {
  const float* x = (const float*)d_in[0];
  const float* w = (const float*)d_in[1];
  float* out = (float*)d_out;
  float* ws = (float*)d_ws;

  // Workspace layout (floats); total ~3.5 MB
  const size_t BNL = (size_t)BB * NN * LL;  // 98304
  const size_t BL = (size_t)BB * LL;        // 384
  const size_t BNF = (size_t)BB * NN * FF;  // 196608
  float* rowsum = ws;
  float* colsum = rowsum + BNL;
  float* diag = colsum + BNL;
  float* trace = diag + BNL;
  float* totsum = trace + BL;
  float* bi2 = totsum + BL;
  float* bjv = bi2 + BNF;
  float* dterm = bjv + BNF;

  reduce_row_kernel<<<BB * NN, 384, 0, stream>>>(x, rowsum, diag);
  reduce_col_kernel<<<BB * NN, 384, 0, stream>>>(x, colsum);
  trace_tot_kernel<<<BB, 64, 0, stream>>>(diag, rowsum, trace, totsum);
  combine_kernel<<<BB * NN, 96, 0, stream>>>(rowsum, colsum, diag, trace,
                                             totsum, w, bi2, bjv, dterm);

  const size_t lds_bytes =
      (size_t)(2 * XS_ELEMS + 2 * WS_ELEMS) * sizeof(float);  // 146432
  dim3 grid(NN / 16, NN / 16, BB);  // (j-tile, i-tile, b)
  main_wmma_kernel<<<grid, 512, lds_bytes, stream>>>(x, w, bi2, bjv, dterm,
                                                     out);
}